// LogicDense_5196910428685
// MI455X (gfx1250) — compile-verified
//
#include <hip/hip_runtime.h>
#include <cstdint>
#include <cstddef>

typedef __attribute__((ext_vector_type(2))) float v2f;
typedef __attribute__((ext_vector_type(8))) float v8f;

#define BATCH   4096
#define IN_DIM  8192
#define OUT_DIM 16384

#define ROWS        4                        // batch rows staged in LDS per workgroup
#define LDS_STRIDE  (IN_DIM + 4)             // +4 floats: keep 16B alignment, shift banks per row
#define LDS_BYTES   (ROWS * LDS_STRIDE * 4)  // 131,136 B -> 2 workgroups per 320 KB WGP
#define MAIN_THREADS 256

// Canonical difflogic op->polynomial table (16 ops x [c0,c1,c2,c3])
__device__ __constant__ float C_TAB[16][4] = {
  {0, 0, 0, 0},  {0, 0, 0, 1},  {0, 1, 0,-1},  {0, 1, 0, 0},
  {0, 0, 1,-1},  {0, 0, 1, 0},  {0, 1, 1,-2},  {0, 1, 1,-1},
  {1,-1,-1, 1},  {1,-1,-1, 2},  {1, 0,-1, 0},  {1, 0,-1, 1},
  {1,-1, 0, 0},  {1,-1, 0, 1},  {1, 0, 0,-1},  {1, 0, 0, 0}
};

// ---------------------------------------------------------------------------
// Kernel A: coef[j][0..3] = softmax(weight[j][0..15]) @ C_TAB  via WMMA f32
// One wave handles 16 neurons: D(16x16) = A(16x4 chunk of p) x B(4x16 of C_TAB)
// accumulated over 4 K-chunks with V_WMMA_F32_16X16X4_F32 (exact fp32).
// ---------------------------------------------------------------------------
__global__ __launch_bounds__(256) void coef_kernel(const float* __restrict__ weight,
                                                   float* __restrict__ coef) {
  const int lane = threadIdx.x & 31;
  const int waveInBlock = threadIdx.x >> 5;
  const int tile = blockIdx.x * 8 + waveInBlock;   // 16 neurons per wave
  const int r    = lane & 15;                      // neuron row within tile / N column
  const bool hi  = lane >= 16;
  const int m    = tile * 16 + r;                  // neuron id (each half-wave redundantly)

  // --- softmax over the 16 logits of neuron m (per lane, fp32) ---
  const float4* wp = (const float4*)(weight + (size_t)m * 16);
  float4 q0 = wp[0], q1 = wp[1], q2 = wp[2], q3 = wp[3];
  float w[16] = { q0.x,q0.y,q0.z,q0.w, q1.x,q1.y,q1.z,q1.w,
                  q2.x,q2.y,q2.z,q2.w, q3.x,q3.y,q3.z,q3.w };
  float mx = w[0];
#pragma unroll
  for (int k = 1; k < 16; ++k) mx = fmaxf(mx, w[k]);
  float p[16]; float s = 0.f;
#pragma unroll
  for (int k = 0; k < 16; ++k) { p[k] = __expf(w[k] - mx); s += p[k]; }
  const float inv = 1.0f / s;
#pragma unroll
  for (int k = 0; k < 16; ++k) p[k] *= inv;

  // --- 4x WMMA f32 16x16x4, K-chunks of the 16 ops ---
  // A (16x4 f32) layout: V0 = {K0 | K2}, V1 = {K1 | K3} across half-waves.
  // B (4x16 f32) layout mirrors A; N = lane&15, only N<4 columns are real.
  const int n = r;
  v8f acc = {};
#pragma unroll
  for (int c = 0; c < 4; ++c) {
    v2f A, B;
    A.x = p[4 * c + (hi ? 2 : 0)];
    A.y = p[4 * c + (hi ? 3 : 1)];
    const int kx = 4 * c + (hi ? 2 : 0);
    const int ky = 4 * c + (hi ? 3 : 1);
    const int nc = n & 3;                       // clamp for safe constant read
    B.x = (n < 4) ? C_TAB[kx][nc] : 0.0f;       // selects, no divergence before WMMA
    B.y = (n < 4) ? C_TAB[ky][nc] : 0.0f;
    acc = __builtin_amdgcn_wmma_f32_16x16x4_f32(
        /*neg_a=*/false, A, /*neg_b=*/false, B,
        /*c_mod=*/(short)0, acc, /*reuse_a=*/false, /*reuse_b=*/false);
  }

  // D layout: VGPR v of lane -> row M = v + (hi?8:0), col N = lane&15.
  if (n < 4) {
#pragma unroll
    for (int v = 0; v < 8; ++v) {
      const int row = v + (hi ? 8 : 0);
      coef[(size_t)(tile * 16 + row) * 4 + n] = acc[v];
    }
  }
}

// ---------------------------------------------------------------------------
// Kernel B: stage ROWS batch rows of x into LDS with async global->LDS DMA,
// then gather + evaluate the per-neuron polynomial. x is read from HBM once;
// 128 KB LDS per WG -> two co-resident WGs per WGP double-buffer the HBM
// read phase against the compute/store phase.
// ---------------------------------------------------------------------------
__global__ __launch_bounds__(MAIN_THREADS) void logic_kernel(
    const float* __restrict__ x,
    const float* __restrict__ coef,
    const int* __restrict__ idx_a,
    const int* __restrict__ idx_b,
    float* __restrict__ out) {
  extern __shared__ float lds[];
  const int t = threadIdx.x;
  const size_t row0 = (size_t)blockIdx.x * ROWS;

  // Async-load ROWS x IN_DIM floats (as float4) into padded LDS rows.
  // 8192 float4 transfers / 256 threads = 32 per thread, fully coalesced.
  for (int i = 0; i < (ROWS * (IN_DIM / 4)) / MAIN_THREADS; ++i) {
    const int q   = t + i * MAIN_THREADS;    // float4 index
    const int row = q >> 11;                 // / (IN_DIM/4)
    const int c4  = q & ((IN_DIM / 4) - 1);
    const float* gp = x + (row0 + row) * IN_DIM + (size_t)c4 * 4;
    const unsigned ldsoff = (unsigned)((row * LDS_STRIDE + c4 * 4) * 4); // bytes
    asm volatile("global_load_async_to_lds_b128 %0, %1, off"
                 :: "v"(ldsoff), "v"(gp)
                 : "memory");
  }
  asm volatile("s_wait_asynccnt 0" ::: "memory");
  __syncthreads();

  // Each thread handles 64 output columns; stores coalesced along j.
  const float4* cf4 = (const float4*)coef;
  for (int j = t; j < OUT_DIM; j += MAIN_THREADS) {
    const int ia = idx_a[j];
    const int ib = idx_b[j];
    const float4 c = cf4[j];
#pragma unroll
    for (int r = 0; r < ROWS; ++r) {
      const float a = lds[r * LDS_STRIDE + ia];
      const float b = lds[r * LDS_STRIDE + ib];
      out[(row0 + r) * OUT_DIM + j] = c.x + c.y * a + c.z * b + c.w * (a * b);
    }
  }
}

// ---------------------------------------------------------------------------
extern "C" void kernel_launch(void* const* d_in, const int* in_sizes, int n_in,
                              void* d_out, int out_size, void* d_ws, size_t ws_size,
                              hipStream_t stream) {
  const float* x      = (const float*)d_in[0];   // (4096, 8192) f32
  const float* weight = (const float*)d_in[1];   // (16384, 16)  f32
  const int*   idx_a  = (const int*)d_in[2];     // (16384,) i32
  const int*   idx_b  = (const int*)d_in[3];     // (16384,) i32
  float*       out    = (float*)d_out;           // (4096, 16384) f32
  float*       coef   = (float*)d_ws;            // (16384, 4) f32 scratch

  // Allow the 128 KB dynamic-LDS allocation per workgroup.
  hipFuncSetAttribute(reinterpret_cast<const void*>(logic_kernel),
                      hipFuncAttributeMaxDynamicSharedMemorySize, LDS_BYTES);

  // Phase 1: per-neuron polynomial coefficients via softmax + WMMA.
  coef_kernel<<<OUT_DIM / 128, 256, 0, stream>>>(weight, coef);

  // Phase 2: streaming gather/evaluate; x read once via async LDS staging,
  // two WGs per WGP overlap HBM reads with compute/stores.
  logic_kernel<<<BATCH / ROWS, MAIN_THREADS, LDS_BYTES, stream>>>(
      x, coef, idx_a, idx_b, out);
}